// Decoder_57586921505036
// MI455X (gfx1250) — compile-verified
//
#include <hip/hip_runtime.h>
#include <hip/hip_bf16.h>

typedef __attribute__((ext_vector_type(2))) float v2f;
typedef __attribute__((ext_vector_type(8))) float v8f;

#define B_    32
#define N_    325
#define E_    64
#define S_    12
#define HOPS_ 3
#define NPAD  336   // 21 tiles of 16 covers N=325

// ---------------------------------------------------------------------------
// Generic batched f32 WMMA GEMM:
//   C[b](M x Ncols) = (accum ? C : 0) + A[b](M x K) * opB[b](K x Ncols) (+ bias[n])
// opB = B row-major (K x Ncols) if btrans==0, else B row-major (Ncols x K) used
// transposed. One wave computes a 32x16 C macro-tile (two stacked 16x16 WMMA
// tiles sharing each B fragment -> 2 v_wmma per 3 b64 loads).
// Out-of-range m/n are CLAMPED for loads (they only feed C entries we never
// store) so the K main loop is branch-free; only the k-tail (K%4) is guarded.
// ---------------------------------------------------------------------------
__global__ void wmma_gemm(const float* __restrict__ A, long long sAb, int lda,
                          const float* __restrict__ Bm, long long sBb, int ldb, int btrans,
                          const float* __restrict__ bias,
                          float* __restrict__ C, long long sCb, int ldc,
                          int M, int Ncols, int K, int accum)
{
    const int lane = threadIdx.x & 31;
    const int wave = threadIdx.x >> 5;
    const int col0 = (blockIdx.y * (blockDim.x >> 5) + wave) * 16;
    const int row0 = blockIdx.x * 32;
    if (col0 >= Ncols) return;  // wave-uniform exit: EXEC stays all-ones per wave

    const float* Ab = A + (long long)blockIdx.z * sAb;
    const float* Bb = Bm + (long long)blockIdx.z * sBb;
    float*       Cb = C + (long long)blockIdx.z * sCb;

    const int m0  = row0 + (lane & 15);
    const int m1  = m0 + 16;
    const int n   = col0 + (lane & 15);
    const int mc0 = (m0 < M) ? m0 : (M - 1);       // clamped: loads always legal
    const int mc1 = (m1 < M) ? m1 : (M - 1);
    const int ncl = (n < Ncols) ? n : (Ncols - 1);
    const int khalf = (lane >> 4) << 1;            // 0 or 2

    const float* Arow0 = Ab + (long long)mc0 * lda + khalf;
    const float* Arow1 = Ab + (long long)mc1 * lda + khalf;
    const int Kmain = K & ~3;

    v8f acc0 = {0.f,0.f,0.f,0.f,0.f,0.f,0.f,0.f};
    v8f acc1 = {0.f,0.f,0.f,0.f,0.f,0.f,0.f,0.f};

    if (btrans) {
        const float* Brow = Bb + (long long)ncl * ldb + khalf;
        #pragma unroll 4
        for (int k0 = 0; k0 < Kmain; k0 += 4) {
            v2f a0, a1, bf;
            a0.x = Arow0[k0];  a0.y = Arow0[k0 + 1];
            a1.x = Arow1[k0];  a1.y = Arow1[k0 + 1];
            bf.x = Brow[k0];   bf.y = Brow[k0 + 1];
            __builtin_prefetch(&Arow0[k0 + 16], 0, 1);
            acc0 = __builtin_amdgcn_wmma_f32_16x16x4_f32(false, a0, false, bf,
                                                         (short)0, acc0, false, false);
            acc1 = __builtin_amdgcn_wmma_f32_16x16x4_f32(false, a1, false, bf,
                                                         (short)0, acc1, false, false);
        }
        if (Kmain < K) {  // single guarded tail chunk
            const int ka = Kmain + khalf;
            v2f a0, a1, bf;
            a0.x = (ka     < K) ? Arow0[Kmain]     : 0.f;
            a0.y = (ka + 1 < K) ? Arow0[Kmain + 1] : 0.f;
            a1.x = (ka     < K) ? Arow1[Kmain]     : 0.f;
            a1.y = (ka + 1 < K) ? Arow1[Kmain + 1] : 0.f;
            bf.x = (ka     < K) ? Brow[Kmain]      : 0.f;
            bf.y = (ka + 1 < K) ? Brow[Kmain + 1]  : 0.f;
            acc0 = __builtin_amdgcn_wmma_f32_16x16x4_f32(false, a0, false, bf,
                                                         (short)0, acc0, false, false);
            acc1 = __builtin_amdgcn_wmma_f32_16x16x4_f32(false, a1, false, bf,
                                                         (short)0, acc1, false, false);
        }
    } else {
        const float* Bcol = Bb + ncl + (long long)khalf * ldb;
        const long long ldb2 = (long long)ldb;
        #pragma unroll 4
        for (int k0 = 0; k0 < Kmain; k0 += 4) {
            v2f a0, a1, bf;
            a0.x = Arow0[k0];  a0.y = Arow0[k0 + 1];
            a1.x = Arow1[k0];  a1.y = Arow1[k0 + 1];
            bf.x = Bcol[(long long)k0 * ldb2];
            bf.y = Bcol[(long long)k0 * ldb2 + ldb2];
            __builtin_prefetch(&Arow0[k0 + 16], 0, 1);
            acc0 = __builtin_amdgcn_wmma_f32_16x16x4_f32(false, a0, false, bf,
                                                         (short)0, acc0, false, false);
            acc1 = __builtin_amdgcn_wmma_f32_16x16x4_f32(false, a1, false, bf,
                                                         (short)0, acc1, false, false);
        }
        if (Kmain < K) {
            const int ka = Kmain + khalf;
            v2f a0, a1, bf;
            a0.x = (ka     < K) ? Arow0[Kmain]     : 0.f;
            a0.y = (ka + 1 < K) ? Arow0[Kmain + 1] : 0.f;
            a1.x = (ka     < K) ? Arow1[Kmain]     : 0.f;
            a1.y = (ka + 1 < K) ? Arow1[Kmain + 1] : 0.f;
            bf.x = (ka     < K) ? Bcol[(long long)Kmain * ldb2]        : 0.f;
            bf.y = (ka + 1 < K) ? Bcol[(long long)Kmain * ldb2 + ldb2] : 0.f;
            acc0 = __builtin_amdgcn_wmma_f32_16x16x4_f32(false, a0, false, bf,
                                                         (short)0, acc0, false, false);
            acc1 = __builtin_amdgcn_wmma_f32_16x16x4_f32(false, a1, false, bf,
                                                         (short)0, acc1, false, false);
        }
    }

    const bool nvld = (n < Ncols);
    const float bv = (bias && nvld) ? bias[n] : 0.f;
    #pragma unroll
    for (int r = 0; r < 8; ++r) {
        const int mr0 = row0 + r + ((lane >> 4) << 3);
        const int mr1 = mr0 + 16;
        if (mr0 < M && nvld) {
            float v = acc0[r] + bv;
            if (accum) Cb[(long long)mr0 * ldc + n] += v;
            else       Cb[(long long)mr0 * ldc + n]  = v;
        }
        if (mr1 < M && nvld) {
            float v = acc1[r] + bv;
            if (accum) Cb[(long long)mr1 * ldc + n] += v;
            else       Cb[(long long)mr1 * ldc + n]  = v;
        }
    }
}

// ---------------------------------------------------------------------------
// Fused  R[b] = softmax_rows( alpha * (relu?) (A[b] @ B[b]^T) )
// A: (M x K) row-major, B: (Ncols x K) row-major. 16-row strip per block;
// energies built with WMMA into LDS, softmax in LDS, single write of result.
// K must be a multiple of 4 here (K=E=64 for all uses) -> branch-free K loop.
// ---------------------------------------------------------------------------
__global__ void wmma_attn_softmax(const float* __restrict__ A, long long sAb, int lda,
                                  const float* __restrict__ Bsrc, long long sBb, int ldb,
                                  float* __restrict__ R, long long sRb, int ldr,
                                  int M, int Ncols, int K, float alpha, int do_relu)
{
    __shared__ float e[16 * NPAD];
    const int lane   = threadIdx.x & 31;
    const int wave   = threadIdx.x >> 5;
    const int nwaves = blockDim.x >> 5;
    const int row0   = blockIdx.x * 16;
    const float* Ab  = A    + (long long)blockIdx.y * sAb;
    const float* Bb  = Bsrc + (long long)blockIdx.y * sBb;

    const int m  = row0 + (lane & 15);
    const int mc = (m < M) ? m : (M - 1);
    const int khalf = (lane >> 4) << 1;
    const float* Arow = Ab + (long long)mc * lda + khalf;
    const int ntiles = (Ncols + 15) >> 4;

    for (int tn = wave; tn < ntiles; tn += nwaves) {
        const int n  = tn * 16 + (lane & 15);
        const int nc = (n < Ncols) ? n : (Ncols - 1);
        const float* Brow = Bb + (long long)nc * ldb + khalf;
        v8f acc = {0.f,0.f,0.f,0.f,0.f,0.f,0.f,0.f};
        #pragma unroll 4
        for (int k0 = 0; k0 < K; k0 += 4) {
            v2f af, bf;
            af.x = Arow[k0];  af.y = Arow[k0 + 1];
            bf.x = Brow[k0];  bf.y = Brow[k0 + 1];
            acc = __builtin_amdgcn_wmma_f32_16x16x4_f32(false, af, false, bf,
                                                        (short)0, acc, false, false);
        }
        #pragma unroll
        for (int r = 0; r < 8; ++r) {
            const int mr = r + ((lane >> 4) << 3);
            float v = acc[r] * alpha;
            if (do_relu) v = fmaxf(v, 0.f);
            e[mr * NPAD + n] = (n < Ncols) ? v : -3.0e38f;
        }
    }
    __syncthreads();
    if ((int)threadIdx.x < 16 && (row0 + (int)threadIdx.x) < M) {
        float* er = &e[threadIdx.x * NPAD];
        float mx = -3.0e38f;
        for (int j = 0; j < Ncols; ++j) mx = fmaxf(mx, er[j]);
        float sum = 0.f;
        for (int j = 0; j < Ncols; ++j) { float ex = __expf(er[j] - mx); er[j] = ex; sum += ex; }
        const float inv = 1.f / sum;
        for (int j = 0; j < Ncols; ++j) er[j] *= inv;
    }
    __syncthreads();
    float* Rb = R + (long long)blockIdx.y * sRb;
    for (int idx = threadIdx.x; idx < 16 * Ncols; idx += blockDim.x) {
        const int mr = idx / Ncols, j = idx - mr * Ncols;
        if (row0 + mr < M) Rb[(long long)(row0 + mr) * ldr + j] = e[mr * NPAD + j];
    }
}

// ---------------------------------------------------------------------------
__global__ void fill_zero(float* p, int n)
{
    int i = blockIdx.x * blockDim.x + threadIdx.x;
    if (i < n) p[i] = 0.f;
}

// GRU gates: gi from scalar prev (OD=1), gh precomputed (rows x 3E)
__global__ void gru_gates(const float* __restrict__ prev, const float* __restrict__ gh,
                          const float* __restrict__ w_ih, const float* __restrict__ b_ih,
                          float* __restrict__ h, float* __restrict__ u, int rows, int E)
{
    int idx = blockIdx.x * blockDim.x + threadIdx.x;
    if (idx >= rows * E) return;
    int r = idx / E, e = idx - r * E;
    float p   = prev[r];
    float i_r = fmaf(p, w_ih[e],         b_ih[e]);
    float i_z = fmaf(p, w_ih[E + e],     b_ih[E + e]);
    float i_n = fmaf(p, w_ih[2 * E + e], b_ih[2 * E + e]);
    long long g0 = (long long)r * 3 * E + e;
    float h_r = gh[g0], h_z = gh[g0 + E], h_n = gh[g0 + 2 * E];
    float rr = 1.f / (1.f + __expf(-(i_r + h_r)));
    float zz = 1.f / (1.f + __expf(-(i_z + h_z)));
    float nn = tanhf(i_n + rr * h_n);
    float hv = h[idx];
    float hnew = (1.f - zz) * nn + zz * hv;
    h[idx] = hnew;
    u[idx] = hnew;
}

// sent[row,hop] = sum_e u*kw / scale   (one wave per row)
__global__ void sent_reduce(const float* __restrict__ u, const float* __restrict__ kw,
                            float* __restrict__ sents, int rows, int E, int hop, float sc)
{
    int row  = blockIdx.x * (blockDim.x >> 5) + (threadIdx.x >> 5);
    int lane = threadIdx.x & 31;
    if (row >= rows) return;
    float s = 0.f;
    for (int e = lane; e < E; e += 32)
        s += u[(long long)row * E + e] * kw[(long long)row * E + e];
    for (int off = 16; off > 0; off >>= 1) s += __shfl_down(s, off);
    if (lane == 0) sents[row * HOPS_ + hop] = s * sc;
}

// BatchNorm1d(num_nodes) training-mode stats: per channel n over (B,E)
__global__ void bn_stats(const float* __restrict__ o, float* __restrict__ mean,
                         float* __restrict__ rstd, int Bn, int Nn, int E)
{
    __shared__ float ss[256], sq[256];
    int n = blockIdx.x;
    float s = 0.f, q = 0.f;
    for (int i = threadIdx.x; i < Bn * E; i += blockDim.x) {
        int b = i / E, e = i - b * E;
        float v = o[((long long)b * Nn + n) * E + e];
        s += v; q += v * v;
    }
    ss[threadIdx.x] = s; sq[threadIdx.x] = q;
    __syncthreads();
    for (int st = 128; st > 0; st >>= 1) {
        if ((int)threadIdx.x < st) {
            ss[threadIdx.x] += ss[threadIdx.x + st];
            sq[threadIdx.x] += sq[threadIdx.x + st];
        }
        __syncthreads();
    }
    if (threadIdx.x == 0) {
        float cnt = (float)(Bn * E);
        float m   = ss[0] / cnt;
        float var = sq[0] / cnt - m * m;
        mean[n] = m;
        rstd[n] = rsqrtf(var + 1e-5f);
    }
}

__global__ void bn_apply(const float* __restrict__ o, const float* __restrict__ mean,
                         const float* __restrict__ rstd, const float* __restrict__ gamma,
                         const float* __restrict__ beta,
                         float* __restrict__ u, float* __restrict__ mems,
                         int rows, int Nn, int E, int hop)
{
    int idx = blockIdx.x * blockDim.x + threadIdx.x;
    if (idx >= rows * E) return;
    int r = idx / E, e = idx - r * E;
    int n = r % Nn;
    float v = (o[idx] - mean[n]) * rstd[n] * gamma[n] + beta[n];
    mems[((long long)r * HOPS_ + hop) * E + e] = v;
    u[idx] += v;
}

// out[b,s,n] = (sum_hop sents*mems) . out_w + out_b ; also feeds prev for next step
__global__ void out_step(const float* __restrict__ sents, const float* __restrict__ mems,
                         const float* __restrict__ ow, const float* __restrict__ ob,
                         float* __restrict__ dout, float* __restrict__ prev,
                         int rows, int Nn, int E, int s, int S)
{
    int row = blockIdx.x * blockDim.x + threadIdx.x;
    if (row >= rows) return;
    float s0 = sents[row * HOPS_ + 0];
    float s1 = sents[row * HOPS_ + 1];
    float s2 = sents[row * HOPS_ + 2];
    const float* m0 = mems + (long long)row * HOPS_ * E;
    const float* m1 = m0 + E;
    const float* m2 = m0 + 2 * E;
    float acc = 0.f;
    for (int e = 0; e < E; ++e)
        acc += (s0 * m0[e] + s1 * m1[e] + s2 * m2[e]) * ow[e];
    acc += ob[0];
    int b = row / Nn, n = row - b * Nn;
    dout[((long long)b * S + s) * Nn + n] = acc;
    prev[row] = acc;
}

// ---------------------------------------------------------------------------
extern "C" void kernel_launch(void* const* d_in, const int* in_sizes, int n_in,
                              void* d_out, int out_size, void* d_ws, size_t ws_size,
                              hipStream_t stream)
{
    (void)in_sizes; (void)n_in; (void)out_size;
    const float* hidden   = (const float*)d_in[0];
    const float* supports = (const float*)d_in[1];
    const float* memory   = (const float*)d_in[3];   // (HOPS+1,B,N,E)
    const float* nodevec1 = (const float*)d_in[4];
    const float* nodevec2 = (const float*)d_in[5];
    const float* gw_ih    = (const float*)d_in[6];
    const float* gw_hh    = (const float*)d_in[7];
    const float* gb_ih    = (const float*)d_in[8];
    const float* gb_hh    = (const float*)d_in[9];
    const float* sent_w   = (const float*)d_in[10];
    const float* bn_gamma = (const float*)d_in[11];
    const float* bn_beta  = (const float*)d_in[12];
    const float* gconv_w  = (const float*)d_in[13];  // (HOPS, 7E, E)
    const float* gconv_b  = (const float*)d_in[14];
    const float* out_w    = (const float*)d_in[15];
    const float* out_b    = (const float*)d_in[16];
    float* dout = (float*)d_out;

    const int BN = B_ * N_, BNE = BN * E_;
    float* w     = (float*)d_ws;
    float* adp   = w;  w += N_ * N_;
    float* score = w;  w += (long long)BN * N_;
    float* h     = w;  w += BNE;
    float* u     = w;  w += BNE;
    float* ghbuf = w;  w += (long long)BN * 3 * E_;
    float* x1    = w;  w += BNE;
    float* x2    = w;  w += BNE;
    float* ybuf  = w;  w += BNE;
    float* obuf  = w;  w += BNE;
    float* kw    = w;  w += BNE;
    float* mems  = w;  w += (long long)BN * HOPS_ * E_;
    float* sents = w;  w += BN * HOPS_;
    float* prevb = w;  w += BN;
    float* meanb = w;  w += N_;
    float* rstdb = w;  w += N_;
    if ((size_t)((char*)w - (char*)d_ws) > ws_size) return;

    const long long sNE = (long long)N_ * E_;
    const long long sNN = (long long)N_ * N_;
    const float inv_scale = 0.125f;  // 1/sqrt(E)
    const int mt_BN = (BN + 31) / 32;   // 32-row macro-tiles
    const int mt_N  = (N_ + 31) / 32;

    // adp = softmax(relu(nodevec1 @ nodevec2^T), axis=-1)   (single batch)
    wmma_attn_softmax<<<dim3((N_ + 15) / 16, 1), 128, 0, stream>>>(
        nodevec1, 0, E_, nodevec2, 0, E_, adp, 0, N_, N_, N_, E_, 1.0f, 1);

    hipMemcpyAsync(h, hidden, (size_t)BNE * sizeof(float),
                   hipMemcpyDeviceToDevice, stream);
    fill_zero<<<(BN + 255) / 256, 256, 0, stream>>>(prevb, BN);

    for (int s = 0; s < S_; ++s) {
        // gh = h @ w_hh^T + b_hh   (BN x 64) * (64 x 192)
        {
            int ntN = (3 * E_ + 15) / 16;
            wmma_gemm<<<dim3(mt_BN, (ntN + 3) / 4, 1), 128, 0, stream>>>(
                h, 0, E_, gw_hh, 0, E_, /*btrans=*/1, gb_hh,
                ghbuf, 0, 3 * E_, BN, 3 * E_, E_, 0);
        }
        gru_gates<<<(BN * E_ + 255) / 256, 256, 0, stream>>>(
            prevb, ghbuf, gw_ih, gb_ih, h, u, BN, E_);

        for (int hop = 0; hop < HOPS_; ++hop) {
            const float* key = memory + (long long)hop * BNE;
            const float* x   = memory + (long long)(hop + 1) * BNE;
            const float* Wh  = gconv_w + (long long)hop * 7 * E_ * E_;

            // score = softmax(u @ key^T / sqrt(E)), batched over B
            wmma_attn_softmax<<<dim3((N_ + 15) / 16, B_), 128, 0, stream>>>(
                u, sNE, E_, key, sNE, E_, score, sNN, N_,
                N_, N_, E_, inv_scale, 0);

            // kw = key @ sent_w[hop]   (flattened BN x 64)
            wmma_gemm<<<dim3(mt_BN, 1, 1), 128, 0, stream>>>(
                key, 0, E_, sent_w + (long long)hop * E_ * E_, 0, E_, 0, nullptr,
                kw, 0, E_, BN, E_, E_, 0);
            sent_reduce<<<(BN + 3) / 4, 128, 0, stream>>>(
                u, kw, sents, BN, E_, hop, inv_scale);

            // o = x @ W[chunk0] + b   (start of the 7-chunk accumulation)
            wmma_gemm<<<dim3(mt_BN, 1, 1), 128, 0, stream>>>(
                x, 0, E_, Wh, 0, E_, 0, gconv_b + hop * E_,
                obuf, 0, E_, BN, E_, E_, 0);

            for (int j = 0; j < 3; ++j) {
                const float* adj = (j < 2) ? (supports + (long long)j * sNN) : adp;
                // x1 = adj @ x   (adj shared across batch)
                wmma_gemm<<<dim3(mt_N, 1, B_), 128, 0, stream>>>(
                    adj, 0, N_, x, sNE, E_, 0, nullptr,
                    x1, sNE, E_, N_, E_, N_, 0);
                // y1 = score @ x1 (batched)
                wmma_gemm<<<dim3(mt_N, 1, B_), 128, 0, stream>>>(
                    score, sNN, N_, x1, sNE, E_, 0, nullptr,
                    ybuf, sNE, E_, N_, E_, N_, 0);
                // o += y1 @ W[1+2j]
                wmma_gemm<<<dim3(mt_BN, 1, 1), 128, 0, stream>>>(
                    ybuf, 0, E_, Wh + (long long)(1 + 2 * j) * E_ * E_, 0, E_, 0, nullptr,
                    obuf, 0, E_, BN, E_, E_, 1);
                // x2 = adj @ x1
                wmma_gemm<<<dim3(mt_N, 1, B_), 128, 0, stream>>>(
                    adj, 0, N_, x1, sNE, E_, 0, nullptr,
                    x2, sNE, E_, N_, E_, N_, 0);
                // y2 = score @ x2
                wmma_gemm<<<dim3(mt_N, 1, B_), 128, 0, stream>>>(
                    score, sNN, N_, x2, sNE, E_, 0, nullptr,
                    ybuf, sNE, E_, N_, E_, N_, 0);
                // o += y2 @ W[2+2j]
                wmma_gemm<<<dim3(mt_BN, 1, 1), 128, 0, stream>>>(
                    ybuf, 0, E_, Wh + (long long)(2 + 2 * j) * E_ * E_, 0, E_, 0, nullptr,
                    obuf, 0, E_, BN, E_, E_, 1);
            }
            bn_stats<<<N_, 256, 0, stream>>>(obuf, meanb, rstdb, B_, N_, E_);
            bn_apply<<<(BN * E_ + 255) / 256, 256, 0, stream>>>(
                obuf, meanb, rstdb, bn_gamma + hop * N_, bn_beta + hop * N_,
                u, mems, BN, N_, E_, hop);
        }
        out_step<<<(BN + 255) / 256, 256, 0, stream>>>(
            sents, mems, out_w, out_b, dout, prevb, BN, N_, E_, s, S_);
    }
}